// LossUnsupervisedEntropy_47674136986050
// MI455X (gfx1250) — compile-verified
//
#include <hip/hip_runtime.h>
#include <math.h>

typedef float v2f __attribute__((ext_vector_type(2)));
typedef float v8f __attribute__((ext_vector_type(8)));

#define DIM 128
#define KC  64
#define LDSS 132   // padded row stride (floats): stride%64==4 -> conflict-free B reads

__global__ void zero_ws_kernel(float* ws) {
    if (threadIdx.x == 0) ws[0] = 0.0f;
}

__global__ __launch_bounds__(256) void intra_entropy_kernel(
        const float* __restrict__ x,
        const float* __restrict__ m,
        float* __restrict__ ws) {
    __shared__ float m_lds[KC * LDSS];
    __shared__ float mnorm2[KC];

    // Stage m (64x128 f32 = 32KB) into LDS, coalesced.
    for (int e = threadIdx.x; e < KC * DIM; e += 256) {
        int k = e >> 7;        // /128
        int d = e & 127;
        m_lds[k * LDSS + d] = m[e];
    }
    __syncthreads();
    if (threadIdx.x < KC) {
        const float* row = &m_lds[threadIdx.x * LDSS];
        float s = 0.0f;
        #pragma unroll 8
        for (int d = 0; d < DIM; ++d) s += row[d] * row[d];
        mnorm2[threadIdx.x] = s;
    }
    __syncthreads();

    const int lane = threadIdx.x & 31;
    const int wave = threadIdx.x >> 5;
    const int half = lane >> 4;     // 0: K=k0,k0+1  1: K=k0+2,k0+3 (A/B frag halves)
    const int l16  = lane & 15;

    // Each wave owns a 16-row tile of x: rows [rowBase, rowBase+16)
    const long rowBase = (long)blockIdx.x * 128 + (long)wave * 16;
    const float* xrow = x + (rowBase + l16) * DIM;   // lane's A row (M = l16)

    v8f acc0 = {}; v8f acc1 = {}; v8f acc2 = {}; v8f acc3 = {};

    #pragma unroll 8
    for (int k0 = 0; k0 < DIM; k0 += 4) {
        const int col = k0 + 2 * half;
        // A fragment: 16x4 f32 tile of x (row M = l16, K = col, col+1)
        v2f a = *(const v2f*)(xrow + col);
        // B fragments: 4x16 f32 tiles of m^T, i.e. m[n][col..col+1], n striped on lanes
        v2f b0 = *(const v2f*)&m_lds[(l16 +  0) * LDSS + col];
        v2f b1 = *(const v2f*)&m_lds[(l16 + 16) * LDSS + col];
        v2f b2 = *(const v2f*)&m_lds[(l16 + 32) * LDSS + col];
        v2f b3 = *(const v2f*)&m_lds[(l16 + 48) * LDSS + col];
        acc0 = __builtin_amdgcn_wmma_f32_16x16x4_f32(false, a, false, b0, (short)0, acc0, false, false);
        acc1 = __builtin_amdgcn_wmma_f32_16x16x4_f32(false, a, false, b1, (short)0, acc1, false, false);
        acc2 = __builtin_amdgcn_wmma_f32_16x16x4_f32(false, a, false, b2, (short)0, acc2, false, false);
        acc3 = __builtin_amdgcn_wmma_f32_16x16x4_f32(false, a, false, b3, (short)0, acc3, false, false);
    }

    // C layout: lane l holds N = (l&15) + 16*j, VGPR slot v holds M = v (lanes 0-15)
    // or M = v+8 (lanes 16-31). Softmax over K=64 = reduce over 16 lanes x 4 accs.
    const float n2_0 = mnorm2[l16 +  0];
    const float n2_1 = mnorm2[l16 + 16];
    const float n2_2 = mnorm2[l16 + 32];
    const float n2_3 = mnorm2[l16 + 48];

    float Hsum = 0.0f;
    #pragma unroll
    for (int v = 0; v < 8; ++v) {
        // score s = 2*x.m - ||m||^2  (||x||^2 cancels in softmax)
        float s0 = 2.0f * acc0[v] - n2_0;
        float s1 = 2.0f * acc1[v] - n2_1;
        float s2 = 2.0f * acc2[v] - n2_2;
        float s3 = 2.0f * acc3[v] - n2_3;

        float mx = fmaxf(fmaxf(s0, s1), fmaxf(s2, s3));
        #pragma unroll
        for (int off = 1; off < 16; off <<= 1)
            mx = fmaxf(mx, __shfl_xor(mx, off, 32));

        float e0 = __expf(s0 - mx);
        float e1 = __expf(s1 - mx);
        float e2 = __expf(s2 - mx);
        float e3 = __expf(s3 - mx);
        float se = e0 + e1 + e2 + e3;
        float ss = s0 * e0 + s1 * e1 + s2 * e2 + s3 * e3;
        #pragma unroll
        for (int off = 1; off < 16; off <<= 1) {
            se += __shfl_xor(se, off, 32);
            ss += __shfl_xor(ss, off, 32);
        }
        // H = lse - sum(p*s) ; lse = mx + log(se)
        float H = (mx + __logf(se)) - ss / se;
        Hsum += H;   // identical across each 16-lane half
    }
    // lanes 0 and 16 each hold entropies of 8 rows of this wave's tile
    if (l16 == 0) atomicAdd(ws, Hsum);
}

__global__ void finalize_kernel(const float* __restrict__ m,
                                const float* __restrict__ ws,
                                float* __restrict__ out, int n_rows) {
    __shared__ float mu[DIM];
    const int t = threadIdx.x;
    if (t < DIM) {
        float s = 0.0f;
        #pragma unroll 8
        for (int k = 0; k < KC; ++k) s += m[k * DIM + t];
        mu[t] = s * (1.0f / (float)KC);
    }
    __syncthreads();
    if (t < 32) {
        float sv[2];
        #pragma unroll
        for (int j = 0; j < 2; ++j) {
            int k = t + 32 * j;
            float dot = 0.0f, n2 = 0.0f;
            #pragma unroll 8
            for (int d = 0; d < DIM; ++d) {
                float mv = m[k * DIM + d];
                dot += mu[d] * mv;
                n2  += mv * mv;
            }
            sv[j] = 2.0f * dot - n2;   // ||mu||^2 cancels in softmax
        }
        float mx = fmaxf(sv[0], sv[1]);
        #pragma unroll
        for (int off = 1; off < 32; off <<= 1)
            mx = fmaxf(mx, __shfl_xor(mx, off, 32));
        float e0 = __expf(sv[0] - mx), e1 = __expf(sv[1] - mx);
        float se = e0 + e1;
        float ss = sv[0] * e0 + sv[1] * e1;
        #pragma unroll
        for (int off = 1; off < 32; off <<= 1) {
            se += __shfl_xor(se, off, 32);
            ss += __shfl_xor(ss, off, 32);
        }
        if (t == 0) {
            float inter = (mx + __logf(se)) - ss / se;
            float intra = ws[0] / (float)n_rows;
            out[0] = intra - inter;   // total (LAMB = 1)
            out[1] = intra;
            out[2] = inter;
        }
    }
}

extern "C" void kernel_launch(void* const* d_in, const int* in_sizes, int n_in,
                              void* d_out, int out_size, void* d_ws, size_t ws_size,
                              hipStream_t stream) {
    const float* x = (const float*)d_in[0];
    const float* m = (const float*)d_in[1];
    float* out = (float*)d_out;
    float* ws  = (float*)d_ws;

    const int n_rows = in_sizes[0] / DIM;      // 262144
    const int blocks = n_rows / 128;           // 16 rows/wave * 8 waves/block

    zero_ws_kernel<<<1, 32, 0, stream>>>(ws);
    intra_entropy_kernel<<<blocks, 256, 0, stream>>>(x, m, ws);
    finalize_kernel<<<1, 128, 0, stream>>>(m, ws, out, n_rows);
}